// LocalAttention_28836410425748
// MI455X (gfx1250) — compile-verified
//
#include <hip/hip_runtime.h>

#define B_    4
#define S_    2048
#define D_    256
#define W_    33
#define HALF_ 16
#define KT_   48   // 3 x 16 key rows covering the +/-16 window of a 16-query tile

typedef __attribute__((ext_vector_type(2))) float v2f;
typedef __attribute__((ext_vector_type(8))) float v8f;

__device__ __forceinline__ int clampS(int x) {
    return x < 0 ? 0 : (x > (S_ - 1) ? (S_ - 1) : x);
}

// D = A(16x4, f32) * B(4x16, f32) + C(16x16, f32)  -- exact f32 matrix op
__device__ __forceinline__ v8f wmma4(v2f a, v2f b, v8f c) {
    return __builtin_amdgcn_wmma_f32_16x16x4_f32(
        /*neg_a=*/false, a, /*neg_b=*/false, b,
        /*c_mod=*/(short)0, c, /*reuse_a=*/false, /*reuse_b=*/false);
}

__global__ __launch_bounds__(128, 1) void LocalAttention_gfx1250_kernel(
    const float* __restrict__ q, const float* __restrict__ kmat,
    const float* __restrict__ vmat, const int* __restrict__ mask,
    float* __restrict__ out_v, float* __restrict__ out_p)
{
    __shared__ float sPart[4][16][KT_];     // per-wave partial scores (k-split)
    __shared__ float sP[16][KT_ + 1];       // softmaxed P, padded stride (49) vs 64 banks

    const int lane = threadIdx.x & 31;
    const int wave = threadIdx.x >> 5;
    const int m    = lane & 15;             // A: row M / B: col N owned by this lane
    const int half = lane >> 4;             // lanes 16-31 hold K+2 / M+8 halves
    const int e2   = 2 * half;

    const int b  = blockIdx.x >> 7;         // 4 batches x 128 tiles
    const int s0 = (blockIdx.x & 127) << 4; // first query row of this tile

    const size_t bOff = (size_t)b * S_ * D_;

    // ---------------- Stage 1: scores[16x48] = Q[16x256] * K^T, k-split over waves ---
    const float* qrow = q    + bOff + (size_t)(s0 + m) * D_;
    const float* kp0  = kmat + bOff + (size_t)clampS(s0 - HALF_ +  0 + m) * D_;
    const float* kp1  = kmat + bOff + (size_t)clampS(s0 - HALF_ + 16 + m) * D_;
    const float* kp2  = kmat + bOff + (size_t)clampS(s0 - HALF_ + 32 + m) * D_;

    v8f acc0 = {}; v8f acc1 = {}; v8f acc2 = {};
    const int dBeg = wave * 64;
    #pragma unroll 4
    for (int d = dBeg; d < dBeg + 64; d += 4) {
        v2f a  = *(const v2f*)(qrow + d + e2);   // A[m][d+2h .. d+2h+1]
        v2f b0 = *(const v2f*)(kp0  + d + e2);   // B[K][N=m] = key_N[d+K]
        v2f b1 = *(const v2f*)(kp1  + d + e2);
        v2f b2 = *(const v2f*)(kp2  + d + e2);
        acc0 = wmma4(a, b0, acc0);
        acc1 = wmma4(a, b1, acc1);
        acc2 = wmma4(a, b2, acc2);
    }
    #pragma unroll
    for (int r = 0; r < 8; ++r) {               // D elem (vgpr r, lane) = [M=r+8h][N=m]
        int mm = r + 8 * half;
        sPart[wave][mm][ 0 + m] = acc0[r];
        sPart[wave][mm][16 + m] = acc1[r];
        sPart[wave][mm][32 + m] = acc2[r];
    }
    __syncthreads();

    // ---------------- Stage 2: mask + softmax, one thread per query row -------------
    if (threadIdx.x < 16) {
        const int  mm = threadIdx.x;
        const int  s  = s0 + mm;
        float x[W_];
        float mx = -3.4e38f;
        #pragma unroll
        for (int w = 0; w < W_; ++w) {
            int   kl  = mm + w;                 // local key col in [mm, mm+32] c [0,47]
            float raw = sPart[0][mm][kl] + sPart[1][mm][kl]
                      + sPart[2][mm][kl] + sPart[3][mm][kl];
            int  j    = s - HALF_ + w;
            bool inb  = (j >= 0) && (j < S_);
            int  mv   = mask[(size_t)b * S_ + clampS(j)];
            float sc  = (inb && (mv != 0)) ? raw * 0.0625f : -1.0e10f;
            x[w] = sc;
            mx   = fmaxf(mx, sc);
        }
        float sum = 0.f;
        #pragma unroll
        for (int w = 0; w < W_; ++w) {
            float t = expf(x[w] - mx);
            x[w] = t;
            sum += t;
        }
        float inv = 1.0f / sum;                 // sum >= 1 (max term contributes 1)
        float* prow = out_p + ((size_t)b * S_ + s) * W_;
        #pragma unroll
        for (int kl = 0; kl < KT_; ++kl) sP[mm][kl] = 0.f;
        #pragma unroll
        for (int w = 0; w < W_; ++w) {
            float p = x[w] * inv;
            prow[w] = p;
            sP[mm][mm + w] = p;                 // banded scatter for dense PV GEMM
        }
    }
    __syncthreads();

    // ---------------- Stage 3: out[16x256] = P[16x48] * V[48x256], n-split over waves
    const float* vbase = vmat + bOff;
    const int dT = wave * 64 + m;               // this lane's output column base
    v8f o0 = {}; v8f o1 = {}; v8f o2 = {}; v8f o3 = {};
    #pragma unroll 4
    for (int k0 = 0; k0 < KT_; k0 += 4) {
        int klA = k0 + e2;
        v2f a = { sP[m][klA], sP[m][klA + 1] };                     // A[m][k]
        const float* vr0 = vbase + (size_t)clampS(s0 - HALF_ + klA    ) * D_ + dT;
        const float* vr1 = vbase + (size_t)clampS(s0 - HALF_ + klA + 1) * D_ + dT;
        v2f bb;
        bb = (v2f){ vr0[ 0], vr1[ 0] }; o0 = wmma4(a, bb, o0);      // B[k][n=dT+16g]
        bb = (v2f){ vr0[16], vr1[16] }; o1 = wmma4(a, bb, o1);
        bb = (v2f){ vr0[32], vr1[32] }; o2 = wmma4(a, bb, o2);
        bb = (v2f){ vr0[48], vr1[48] }; o3 = wmma4(a, bb, o3);
    }
    #pragma unroll
    for (int r = 0; r < 8; ++r) {
        size_t row  = (size_t)b * S_ + s0 + r + 8 * half;           // M = r + 8h
        float* orow = out_v + row * D_ + wave * 64 + m;
        orow[ 0] = o0[r];
        orow[16] = o1[r];
        orow[32] = o2[r];
        orow[48] = o3[r];
    }
}

extern "C" void kernel_launch(void* const* d_in, const int* in_sizes, int n_in,
                              void* d_out, int out_size, void* d_ws, size_t ws_size,
                              hipStream_t stream) {
    (void)in_sizes; (void)n_in; (void)out_size; (void)d_ws; (void)ws_size;
    const float* q    = (const float*)d_in[0];
    const float* k    = (const float*)d_in[1];
    const float* v    = (const float*)d_in[2];
    const int*   mask = (const int*)d_in[3];
    float* out_v = (float*)d_out;
    float* out_p = out_v + (size_t)B_ * S_ * D_;   // tuple order: values, p_attn
    dim3 grid(B_ * (S_ / 16));                     // 512 blocks, 4 waves each
    LocalAttention_gfx1250_kernel<<<grid, 128, 0, stream>>>(q, k, v, mask, out_v, out_p);
}